// AdaptiveSpatioTemporalFusion_36601711296537
// MI455X (gfx1250) — compile-verified
//
#include <hip/hip_runtime.h>

// ============================================================================
// AdaptiveSpatioTemporalFusion for MI455X (gfx1250, wave32, WMMA).
// All convs / deformable convs / adaptive convs -> im2col + fp32 GEMM on
// V_WMMA_F32_16X16X4_F32.  GEMM uses 4-wave blocks: B (im2col) slab staged
// through LDS and shared by 4 waves; each wave register-blocks 2x2 tiles so
// every fragment feeds two WMMAs.  Main K-slabs are branchless (pointer
// increment + mask multiplies); only the final ragged-K slab is guarded.
// Workspace arena: ~88 MB of d_ws.
// ============================================================================

#define BN    2
#define Hs    48
#define Wsz   48
#define HW    (48*48)
#define INC   256
#define DIMC  64
#define CORRC 49
#define ICWC  177      /* 2*DIMC + CORRC */
#define KKC   9

typedef float v2f __attribute__((ext_vector_type(2)));
typedef float v8f __attribute__((ext_vector_type(8)));

// ---------------------------------------------------------------------------
// GEMM:  C[N][M] = W[N][K] * Bc[K][M]   (+bias, optional ReLU)
// Block = 128 threads (4 waves).  Macro tile: 32 pixels (M) x 128 channels (N).
// Wave w owns rows [nb, nb+32), nb = blockIdx.y*128 + w*32, as two 16-row
// WMMA tiles; the two 16-pixel column tiles are shared block-wide via LDS.
// B slab (16 K x 32 M) stored pair-interleaved ([k/2][m][k&1]) so a B
// fragment is one aligned 8B ds_load.  A fragment (fp32 16x4, ISA 7.12.2):
// lane l row = l%16, VGPR0 <-> K=2*(l>>4), VGPR1 <-> K=2*(l>>4)+1.
// EXEC all-ones through the WMMA region (masks are selects/multiplies).
// ---------------------------------------------------------------------------
__global__ __launch_bounds__(128)
void k_gemm_wmma(const float* __restrict__ Wm, const float* __restrict__ Bc,
                 float* __restrict__ Cc, int Mdim, int Ndim, int Kdim,
                 const float* __restrict__ bias, int relu)
{
    __shared__ float sB[8][32][2];          // 16 K x 32 M, pair-interleaved

    const int tid  = threadIdx.x;
    const int wave = tid >> 5;
    const int lane = tid & 31;
    const int half = lane >> 4;             // 0/1
    const int lm   = lane & 15;

    const int mbase = blockIdx.x * 32;
    const int nb    = blockIdx.y * 128 + wave * 32;   // this wave's N base

    // ---- staging thread mapping: fixed column mm, k-in-slab = kloc + 4r ----
    const int kloc = tid >> 5;              // 0..3
    const int mm   = tid & 31;
    const int gm   = mbase + mm;
    const int gmc  = (gm < Mdim) ? gm : (Mdim - 1);
    const float stm = (gm < Mdim) ? 1.f : 0.f;
    const long rstep = (long)4  * Mdim;     // k advances 4 between r's
    const long kstep = (long)16 * Mdim;     // one slab
    const float* sp = Bc + gmc + (long)kloc * Mdim;     // clamped, in-bounds
    float* const sdst = &sB[kloc >> 1][mm][kloc & 1];   // +r*128 floats per r

    // ---- per-wave fragment rows/columns ----
    const int m0 = mbase + lm, m1 = m0 + 16;
    const int n0 = nb + lm,    n1 = n0 + 16;
    const float am0 = (n0 < Ndim) ? 1.f : 0.f;
    const float am1 = (n1 < Ndim) ? 1.f : 0.f;
    const float* __restrict__ Wr0 = Wm + (long)((n0 < Ndim) ? n0 : (Ndim - 1)) * Kdim;
    const float* __restrict__ Wr1 = Wm + (long)((n1 < Ndim) ? n1 : (Ndim - 1)) * Kdim;

    v8f acc00 = {}, acc01 = {}, acc10 = {}, acc11 = {};

    const int Kslab = Kdim & ~15;           // full 16-K slabs (K>=256 always)
    int k0 = 0;
    for (; k0 < Kslab; k0 += 16) {
        // ---- branchless staging: 4 coalesced loads + 4 ds_stores ----
        #pragma unroll
        for (int r = 0; r < 4; ++r) {
            const float v = sp[r * rstep] * stm;
            __builtin_prefetch(sp + r * rstep + kstep, 0, 1);   // next slab
            sdst[r * 128] = v;
        }
        sp += kstep;
        __syncthreads();

        // ---- 4 WMMA K-steps, 2x2 register blocking, unguarded A loads ----
        #pragma unroll
        for (int kk = 0; kk < 16; kk += 4) {
            const int ka = kk + (half << 1);
            const int kg = k0 + ka;
            const v2f b0 = *reinterpret_cast<const v2f*>(&sB[ka >> 1][lm][0]);
            const v2f b1 = *reinterpret_cast<const v2f*>(&sB[ka >> 1][lm + 16][0]);
            v2f a0, a1;
            a0.x = Wr0[kg]     * am0;
            a0.y = Wr0[kg + 1] * am0;
            a1.x = Wr1[kg]     * am1;
            a1.y = Wr1[kg + 1] * am1;
            acc00 = __builtin_amdgcn_wmma_f32_16x16x4_f32(false, a0, false, b0,
                                                          (short)0, acc00, false, false);
            acc01 = __builtin_amdgcn_wmma_f32_16x16x4_f32(false, a0, false, b1,
                                                          (short)0, acc01, false, false);
            acc10 = __builtin_amdgcn_wmma_f32_16x16x4_f32(false, a1, false, b0,
                                                          (short)0, acc10, false, false);
            acc11 = __builtin_amdgcn_wmma_f32_16x16x4_f32(false, a1, false, b1,
                                                          (short)0, acc11, false, false);
        }
        __syncthreads();
    }

    if (k0 < Kdim) {    // ragged-K tail slab (K=4425,1593): uniform branch
        #pragma unroll
        for (int r = 0; r < 4; ++r) {
            const int gk = k0 + kloc + 4 * r;
            const float* q = Bc + (long)((gk < Kdim) ? gk : (Kdim - 1)) * Mdim + gmc;
            sdst[r * 128] = (gk < Kdim) ? (*q) * stm : 0.f;
        }
        __syncthreads();
        #pragma unroll
        for (int kk = 0; kk < 16; kk += 4) {
            const int ka = kk + (half << 1);
            const int kg = k0 + ka;
            const v2f b0 = *reinterpret_cast<const v2f*>(&sB[ka >> 1][lm][0]);
            const v2f b1 = *reinterpret_cast<const v2f*>(&sB[ka >> 1][lm + 16][0]);
            v2f a0, a1;
            a0.x = (kg     < Kdim) ? Wr0[kg]     * am0 : 0.f;
            a0.y = (kg + 1 < Kdim) ? Wr0[kg + 1] * am0 : 0.f;
            a1.x = (kg     < Kdim) ? Wr1[kg]     * am1 : 0.f;
            a1.y = (kg + 1 < Kdim) ? Wr1[kg + 1] * am1 : 0.f;
            acc00 = __builtin_amdgcn_wmma_f32_16x16x4_f32(false, a0, false, b0,
                                                          (short)0, acc00, false, false);
            acc01 = __builtin_amdgcn_wmma_f32_16x16x4_f32(false, a0, false, b1,
                                                          (short)0, acc01, false, false);
            acc10 = __builtin_amdgcn_wmma_f32_16x16x4_f32(false, a1, false, b0,
                                                          (short)0, acc10, false, false);
            acc11 = __builtin_amdgcn_wmma_f32_16x16x4_f32(false, a1, false, b1,
                                                          (short)0, acc11, false, false);
        }
    }

    // ---- store D: VGPR r, lane half h -> row (ntile + r + 8h), col m ----
    auto store = [&](const v8f& acc, int ntile, int m) {
        if (m < Mdim) {
            #pragma unroll
            for (int r = 0; r < 8; ++r) {
                const int nn = ntile + r + (half << 3);
                if (nn < Ndim) {
                    float v = acc[r];
                    if (bias) v += bias[nn];
                    if (relu && v < 0.f) v = 0.f;
                    Cc[(long)nn * Mdim + m] = v;
                }
            }
        }
    };
    store(acc00, nb,      m0);
    store(acc01, nb,      m1);
    store(acc10, nb + 16, m0);
    store(acc11, nb + 16, m1);
}

// ---------------------------------------------------------------------------
// im2col: X[C][H][W] -> cols[(c*KH*KW + kh*KW + kw)][oh*OW + ow], zero pad.
// ---------------------------------------------------------------------------
__global__ void k_im2col(const float* __restrict__ X, float* __restrict__ cols,
                         int Cin, int H, int W, int KH, int KW,
                         int stride, int pad, int OH, int OW)
{
    long idx = (long)blockIdx.x * blockDim.x + threadIdx.x;
    const int M = OH * OW;
    const long total = (long)Cin * KH * KW * M;
    if (idx >= total) return;
    int m = (int)(idx % M);
    long t = idx / M;
    int kw = (int)(t % KW); t /= KW;
    int kh = (int)(t % KH); t /= KH;
    int c  = (int)t;
    int ow = m % OW, oh = m / OW;
    int ih = oh * stride - pad + kh;
    int iw = ow * stride - pad + kw;
    float v = 0.f;
    if (ih >= 0 && ih < H && iw >= 0 && iw < W)
        v = X[((long)c * H + ih) * W + iw];
    cols[idx] = v;
}

// ---------------------------------------------------------------------------
// deform-im2col: bilinear-sampled im2col (3x3, pad 1).
// off: [KK][2][H][W]  (channel q*2+0 = dy, q*2+1 = dx)
// cols[(c*9 + q)][h*W + w]
// ---------------------------------------------------------------------------
__global__ void k_deform_im2col(const float* __restrict__ X,
                                const float* __restrict__ off,
                                float* __restrict__ cols,
                                int Cin, int H, int W)
{
    long idx = (long)blockIdx.x * blockDim.x + threadIdx.x;
    const int M = H * W;
    const long total = (long)Cin * KKC * M;
    if (idx >= total) return;
    int m = (int)(idx % M);
    long t = idx / M;
    int q = (int)(t % KKC);
    int c = (int)(t / KKC);
    int w = m % W, h = m / W;
    int ki = q / 3, kj = q % 3;
    float oy = off[((long)(q * 2 + 0) * H + h) * W + w];
    float ox = off[((long)(q * 2 + 1) * H + h) * W + w];
    float py = (float)(h - 1 + ki) + oy;
    float px = (float)(w - 1 + kj) + ox;
    float y0f = floorf(py), x0f = floorf(px);
    int   y0 = (int)y0f,    x0 = (int)x0f;
    float ay = py - y0f,    ax = px - x0f;
    const float* Xc = X + (long)c * M;
    float v00 = (y0   >= 0 && y0   < H && x0   >= 0 && x0   < W) ? Xc[y0 * W + x0]           : 0.f;
    float v01 = (y0   >= 0 && y0   < H && x0+1 >= 0 && x0+1 < W) ? Xc[y0 * W + x0 + 1]       : 0.f;
    float v10 = (y0+1 >= 0 && y0+1 < H && x0   >= 0 && x0   < W) ? Xc[(y0 + 1) * W + x0]     : 0.f;
    float v11 = (y0+1 >= 0 && y0+1 < H && x0+1 >= 0 && x0+1 < W) ? Xc[(y0 + 1) * W + x0 + 1] : 0.f;
    cols[idx] = v00 * (1.f - ay) * (1.f - ax) + v01 * (1.f - ay) * ax
              + v10 * ay * (1.f - ax)         + v11 * ay * ax;
}

// ---------------------------------------------------------------------------
// correlation (one batch): out[d][h][w] = mean_c A[c][h][w]*B[c][h+dy][w+dx],
// d = dyi*7 + dxi, dy = 2*(dyi-3); zero outside.
// ---------------------------------------------------------------------------
__global__ void k_corr(const float* __restrict__ A, const float* __restrict__ Bt,
                       float* __restrict__ out, int C, int H, int W)
{
    int idx = blockIdx.x * blockDim.x + threadIdx.x;
    const int M = H * W;
    if (idx >= CORRC * M) return;
    int m = idx % M, d = idx / M;
    int dy = 2 * ((d / 7) - 3), dx = 2 * ((d % 7) - 3);
    int h = m / W, w = m % W;
    int hh = h + dy, ww = w + dx;
    float s = 0.f;
    if (hh >= 0 && hh < H && ww >= 0 && ww < W) {
        const float* ap = A + m;
        const float* bp = Bt + hh * W + ww;
        for (int c = 0; c < C; ++c) s += ap[(long)c * M] * bp[(long)c * M];
    }
    out[idx] = s * (1.f / (float)C);
}

// mean over spatial M:  fw[b*256 + c] = mean_m X[b][c][m]
__global__ void k_mean_hw(const float* __restrict__ X, float* __restrict__ fw,
                          int C, int M)
{
    int i = blockIdx.x * blockDim.x + threadIdx.x;
    if (i >= BN * C) return;
    int b = i / C, c = i % C;
    const float* p = X + ((long)b * C + c) * M;
    float s = 0.f;
    for (int m = 0; m < M; ++m) s += p[m];
    fw[b * 256 + c] = s / (float)M;
}

// grouped 1x1: wout[b][o][k] = fw[b][o]*w[o][k] + bb[o][k]
__global__ void k_make_weight(const float* __restrict__ fw, const float* __restrict__ w,
                              const float* __restrict__ bb, float* __restrict__ wout,
                              int O, int Kc, long ostride)
{
    long i = (long)blockIdx.x * blockDim.x + threadIdx.x;
    const long total = (long)BN * O * Kc;
    if (i >= total) return;
    long t = i;
    int k = (int)(t % Kc); t /= Kc;
    int o = (int)(t % O);
    int b = (int)(t / O);
    wout[b * ostride + (long)o * Kc + k] = fw[b * 256 + o] * w[(long)o * Kc + k]
                                         + bb[(long)o * Kc + k];
}

// final cosine-softmax blend (sw* have 1 channel)
__global__ void k_fuse(const float* __restrict__ d0, const float* __restrict__ d1,
                       const float* __restrict__ sw0, const float* __restrict__ sw1,
                       float* __restrict__ out)
{
    long i = (long)blockIdx.x * blockDim.x + threadIdx.x;
    const long total = (long)BN * INC * HW;
    if (i >= total) return;
    int hw = (int)(i % HW);
    int b  = (int)(i / ((long)INC * HW));
    float s0 = sw0[b * HW + hw], s1 = sw1[b * HW + hw];
    const float eps = 1e-8f;
    float na = fmaxf(fabsf(s0), eps), nb = fmaxf(fabsf(s1), eps);
    float Wx = (s0 * s1) / (na * nb);
    float Wy = (s1 * s1) / (nb * nb);
    float mx = fmaxf(Wx, Wy);
    float e0 = expf(Wx - mx), e1 = expf(Wy - mx);
    float inv = 1.f / (e0 + e1);
    out[i] = d0[i] * (e0 * inv) + d1[i] * (e1 * inv);
}

// ============================================================================
// Host orchestration
// ============================================================================
namespace {

struct TP { const float* p; long bs; };   // base + per-batch stride (floats)

inline int gdim(long total, int bs) { return (int)((total + bs - 1) / bs); }

void gemm(hipStream_t s, const float* W, const float* Bc, float* C,
          int M, int N, int K, const float* bias, int relu)
{
    dim3 g((M + 31) / 32, (N + 127) / 128, 1);
    k_gemm_wmma<<<g, 128, 0, s>>>(W, Bc, C, M, N, K, bias, relu);
}

void im2col(hipStream_t s, const float* X, float* cols, int C, int H, int W,
            int KH, int KW, int stride, int pad, int OH, int OW)
{
    long total = (long)C * KH * KW * OH * OW;
    k_im2col<<<gdim(total, 256), 256, 0, s>>>(X, cols, C, H, W, KH, KW, stride, pad, OH, OW);
}

// standard conv for all batches
void conv_all(hipStream_t s, TP x, const float* w, float* out, float* cols,
              int Cin, int Cout, int H, int W, int KH, int KW, int stride, int pad,
              const float* bias, int relu)
{
    int OH = (H + 2 * pad - KH) / stride + 1;
    int OW = (W + 2 * pad - KW) / stride + 1;
    int M = OH * OW, K = Cin * KH * KW;
    for (int b = 0; b < BN; ++b) {
        im2col(s, x.p + (long)b * x.bs, cols, Cin, H, W, KH, KW, stride, pad, OH, OW);
        gemm(s, w, cols, out + (long)b * Cout * M, M, Cout, K, nullptr, relu);
    }
}

// deformable conv (3x3 pad1) for all batches; wbstride=0 -> shared weights
void deform_conv_all(hipStream_t s, TP x, const float* off, const float* w,
                     long wbstride, float* out, float* cols, int Cin, int Cout)
{
    const int M = HW, K = Cin * KKC;
    for (int b = 0; b < BN; ++b) {
        long total = (long)Cin * KKC * HW;
        k_deform_im2col<<<gdim(total, 256), 256, 0, s>>>(
            x.p + (long)b * x.bs, off + (long)b * 18 * HW, cols, Cin, Hs, Wsz);
        gemm(s, w + (long)b * wbstride, cols, out + (long)b * Cout * M,
             M, Cout, K, nullptr, 0);
    }
}

struct Bufs {
    float *FEATA, *FEATB, *OFFB, *COLS, *FEATW, *FEATW2;
    float *WB1, *WB2, *WB3, *FW, *WADP, *SN1, *SN2;
};
struct Wts {
    const float *offw[4], *defw[3];
    const float *w0a, *w0b, *w0c, *w1a, *w1b, *w1c;
    const float *wx_w, *wx_b, *wxf_w, *wxf_b;
    const float *s1, *s2, *s3;
};

// stsn_offset(xa, xb) -> final 18-ch offsets in B.OFFB
void stsn_offset(hipStream_t s, const Bufs& B, const Wts& W, TP xa, TP xb)
{
    for (int b = 0; b < BN; ++b) {
        hipMemcpyAsync(B.FEATA + ((long)b * 512) * HW, xa.p + (long)b * xa.bs,
                       sizeof(float) * INC * HW, hipMemcpyDeviceToDevice, s);
        hipMemcpyAsync(B.FEATA + ((long)b * 512 + INC) * HW, xb.p + (long)b * xb.bs,
                       sizeof(float) * INC * HW, hipMemcpyDeviceToDevice, s);
    }
    float* fa = B.FEATA; float* fb = B.FEATB;
    TP tfa{fa, 512L * HW};
    for (int i = 0; i < 3; ++i) {
        tfa.p = fa;
        conv_all(s, tfa, W.offw[i], B.OFFB, B.COLS, 512, 18, Hs, Wsz, 3, 3, 1, 1, nullptr, 0);
        deform_conv_all(s, tfa, B.OFFB, W.defw[i], 0, fb, B.COLS, 512, 512);
        float* t = fa; fa = fb; fb = t;
    }
    tfa.p = fa;
    conv_all(s, tfa, W.offw[3], B.OFFB, B.COLS, 512, 18, Hs, Wsz, 3, 3, 1, 1, nullptr, 0);
}

// astsn_weight -> per-batch (256 x 2304) adaptive weights in B.WADP
void astsn_weight(hipStream_t s, const Bufs& B, const Wts& W,
                  TP c0, TP c1, TP we0, TP we1)
{
    const long ADPS = (long)INC * INC * KKC;
    // featW = concat(corr, x_enc, y_enc)
    for (int b = 0; b < BN; ++b) {
        k_corr<<<gdim((long)CORRC * HW, 256), 256, 0, s>>>(
            c0.p + (long)b * c0.bs, c1.p + (long)b * c1.bs,
            B.FEATW + (long)b * ICWC * HW, DIMC, Hs, Wsz);
        hipMemcpyAsync(B.FEATW + ((long)b * ICWC + CORRC) * HW, we0.p + (long)b * we0.bs,
                       sizeof(float) * DIMC * HW, hipMemcpyDeviceToDevice, s);
        hipMemcpyAsync(B.FEATW + ((long)b * ICWC + CORRC + DIMC) * HW, we1.p + (long)b * we1.bs,
                       sizeof(float) * DIMC * HW, hipMemcpyDeviceToDevice, s);
    }
    TP tfw{B.FEATW, (long)ICWC * HW};
    TP twb1{B.WB1, 64L * 576}, twb2{B.WB2, 64L * 144};
    // weight branch 0
    conv_all(s, tfw,  W.w0a, B.WB1, B.COLS, ICWC, 64,  Hs, Wsz, 5, 5, 2, 2, nullptr, 1);
    conv_all(s, twb1, W.w0b, B.WB2, B.COLS, 64,   64,  24, 24,  5, 5, 2, 2, nullptr, 1);
    conv_all(s, twb2, W.w0c, B.WB3, B.COLS, 64,   ICWC,12, 12,  3, 3, 2, 1, nullptr, 0);
    k_mean_hw<<<gdim(BN * ICWC, 256), 256, 0, s>>>(B.WB3, B.FW, ICWC, 36);
    k_make_weight<<<gdim((long)BN * ICWC * ICWC * KKC, 256), 256, 0, s>>>(
        B.FW, W.wx_w, W.wx_b, B.WADP, ICWC, ICWC * KKC, ADPS);
    // adaptive conv 3x3 pad1 (per-batch weights), ReLU
    for (int b = 0; b < BN; ++b) {
        im2col(s, B.FEATW + (long)b * ICWC * HW, B.COLS, ICWC, Hs, Wsz, 3, 3, 1, 1, Hs, Wsz);
        gemm(s, B.WADP + b * ADPS, B.COLS, B.FEATW2 + (long)b * ICWC * HW,
             HW, ICWC, ICWC * KKC, nullptr, 1);
    }
    // weight branch 1
    TP tfw2{B.FEATW2, (long)ICWC * HW};
    conv_all(s, tfw2, W.w1a, B.WB1, B.COLS, ICWC, 64,  Hs, Wsz, 5, 5, 2, 2, nullptr, 1);
    conv_all(s, twb1, W.w1b, B.WB2, B.COLS, 64,   64,  24, 24,  5, 5, 2, 2, nullptr, 1);
    conv_all(s, twb2, W.w1c, B.WB3, B.COLS, 64,   INC, 12, 12,  3, 3, 2, 1, nullptr, 0);
    k_mean_hw<<<gdim(BN * INC, 256), 256, 0, s>>>(B.WB3, B.FW, INC, 36);
    k_make_weight<<<gdim((long)BN * INC * INC * KKC, 256), 256, 0, s>>>(
        B.FW, W.wxf_w, W.wxf_b, B.WADP, INC, INC * KKC, ADPS);
}

// one full pass -> def_out [BN][256][HW], sw_out [BN][HW]
void run_pass(hipStream_t s, const Bufs& B, const Wts& W,
              TP xin, TP sa, TP sb, TP c0, TP c1, TP we0, TP we1,
              float* def_out, float* sw_out)
{
    const long ADPS = (long)INC * INC * KKC;
    astsn_weight(s, B, W, c0, c1, we0, we1);      // -> B.WADP (persists)
    stsn_offset(s, B, W, sa, sb);                 // -> B.OFFB
    deform_conv_all(s, xin, B.OFFB, B.WADP, ADPS, def_out, B.COLS, INC, INC);
    // s_net
    for (int b = 0; b < BN; ++b) {
        const float* dp = def_out + (long)b * INC * HW;
        im2col(s, dp,    B.COLS, INC, Hs, Wsz, 3, 3, 1, 1, Hs, Wsz);
        gemm(s, W.s1, B.COLS, B.SN1, HW, 128, INC * 9, nullptr, 1);
        im2col(s, B.SN1, B.COLS, 128, Hs, Wsz, 3, 3, 1, 1, Hs, Wsz);
        gemm(s, W.s2, B.COLS, B.SN2, HW, 64, 128 * 9, nullptr, 1);
        im2col(s, B.SN2, B.COLS, 64,  Hs, Wsz, 3, 3, 1, 1, Hs, Wsz);
        gemm(s, W.s3, B.COLS, sw_out + (long)b * HW, HW, 1, 64 * 9, nullptr, 1);
    }
}

} // namespace

extern "C" void kernel_launch(void* const* d_in, const int* in_sizes, int n_in,
                              void* d_out, int out_size, void* d_ws, size_t ws_size,
                              hipStream_t stream)
{
    (void)in_sizes; (void)n_in; (void)out_size; (void)ws_size;
    const float* R0     = (const float*)d_in[0];
    const float* T0     = (const float*)d_in[1];
    const float* inp    = (const float*)d_in[2];
    const float* enc0_w = (const float*)d_in[3];
    const float* enc0_b = (const float*)d_in[4];
    const float* enc1_w = (const float*)d_in[5];
    const float* enc1_b = (const float*)d_in[6];
    Wts W;
    for (int i = 0; i < 4; ++i) W.offw[i] = (const float*)d_in[7 + i];
    for (int i = 0; i < 3; ++i) W.defw[i] = (const float*)d_in[11 + i];
    W.w0a  = (const float*)d_in[14]; W.w0b = (const float*)d_in[15];
    W.w0c  = (const float*)d_in[16]; W.w1a = (const float*)d_in[17];
    W.w1b  = (const float*)d_in[18]; W.w1c = (const float*)d_in[19];
    W.wx_w = (const float*)d_in[20]; W.wx_b = (const float*)d_in[21];
    W.wxf_w= (const float*)d_in[22]; W.wxf_b= (const float*)d_in[23];
    W.s1   = (const float*)d_in[24]; W.s2 = (const float*)d_in[25];
    W.s3   = (const float*)d_in[26];

    // ---- workspace arena (floats), ~88 MB total ----
    float* ws = (float*)d_ws;
    size_t o = 0;
    auto A = [&](long n) { float* r = ws + o; o += (size_t)((n + 63) & ~63L); return r; };
    float* XENC = A((long)BN * DIMC * HW);
    float* YENC = A((long)BN * DIMC * HW);
    float* DEF0 = A((long)BN * INC * HW);
    float* DEF1 = A((long)BN * INC * HW);
    float* SW0  = A((long)BN * HW);
    float* SW1  = A((long)BN * HW);
    Bufs B;
    B.FEATA  = A((long)BN * 512 * HW);
    B.FEATB  = A((long)BN * 512 * HW);
    B.OFFB   = A((long)BN * 18 * HW);
    B.COLS   = A(512L * KKC * HW);            // per-batch im2col staging (max K*M)
    B.FEATW  = A((long)BN * ICWC * HW);
    B.FEATW2 = A((long)BN * ICWC * HW);
    B.WB1    = A((long)BN * 64 * 576);
    B.WB2    = A((long)BN * 64 * 144);
    B.WB3    = A((long)BN * 256 * 36);
    B.FW     = A((long)BN * 256);
    B.WADP   = A((long)BN * INC * INC * KKC); // per-batch adaptive weights
    B.SN1    = A(128L * HW);
    B.SN2    = A(64L * HW);

    // tensor views (base pointer + per-batch stride)
    TP x   {inp,                       2L * INC * HW};
    TP y   {inp + (long)INC * HW,      2L * INC * HW};
    TP xe  {XENC,                      (long)DIMC * HW};
    TP ye  {YENC,                      (long)DIMC * HW};
    TP Rpre{R0,                        2L * DIMC * HW};
    TP Rcur{R0 + (long)DIMC * HW,      2L * DIMC * HW};
    TP Tcur{T0 + (long)DIMC * HW,      2L * DIMC * HW};

    // encoders: 1x1 convs == GEMM with input as [Cin][HW] columns
    for (int b = 0; b < BN; ++b) {
        gemm(stream, enc0_w, x.p + (long)b * x.bs, XENC + (long)b * DIMC * HW,
             HW, DIMC, INC, enc0_b, 0);
        gemm(stream, enc1_w, y.p + (long)b * y.bs, YENC + (long)b * DIMC * HW,
             HW, DIMC, INC, enc1_b, 0);
    }

    // pass 0: deform0 from x with offsets(x,y) and weights(R_pre,T_cur,xe,ye)
    run_pass(stream, B, W, x, x, y, Rpre, Tcur, xe, ye, DEF0, SW0);
    // pass 1: deform1 from y with offsets(y,y) and weights(R_cur,T_cur,ye,ye)
    run_pass(stream, B, W, y, y, y, Rcur, Tcur, ye, ye, DEF1, SW1);

    // cosine-softmax blend -> d_out [BN][256][HW]
    k_fuse<<<gdim((long)BN * INC * HW, 256), 256, 0, stream>>>(
        DEF0, DEF1, SW0, SW1, (float*)d_out);
}